// AlphaLayer_34522947125626
// MI455X (gfx1250) — compile-verified
//
#include <hip/hip_runtime.h>

// Batched Thomas tridiagonal solve, fused fwd+bwd, MI455X (gfx1250).
//   a_i = alpha[i-1]^2, b_i = 1 + alpha[i]^3, c_i = alpha[i+1]^2 + 2*alpha[i+1]
//   fwd: denom = b - a*cp';  cp = c/denom;  dp = (f - a*dp')/denom
//   bwd: u_i = dp_i - cp_i * u_{i+1}
//
// One wave per 32 rows (lane = row). alpha is staged through LDS via the
// Tensor Data Mover (2-D strided tile, LDS padding for bank-conflict-free
// reads), double buffered and synchronized with s_wait_tensorcnt.
// cp/dp intermediates (128 MB) are stored transposed ([i][row], coalesced
// float2) so they stay resident in the 192 MB L2; output is written with
// non-temporal stores so the u stream does not evict them.
// Workspace requirement: ws_size >= B*N*8 bytes (128 MB for B=2048,N=8192).

typedef unsigned int       u32;
typedef unsigned long long u64;
typedef __attribute__((ext_vector_type(4))) u32 v4u;
typedef __attribute__((ext_vector_type(4))) int v4i;
typedef __attribute__((ext_vector_type(8))) int v8i;

#define TILE        32
#define LDS_STRIDE  33                   // 32 dwords + 1 pad dword (TDM pad)
#define TILE_ELEMS  (TILE * LDS_STRIDE)  // floats per LDS tile buffer

// DMA one 32(rows) x 32(cols) f32 tile (row stride = n_stride elements in
// global memory) into LDS at lds_ptr, with 1-dword padding every 32 dwords.
__device__ __forceinline__ void tdm_load_tile(float* lds_ptr,
                                              const float* __restrict__ gptr,
                                              int n_stride, int lane) {
#if defined(__has_builtin) && __has_builtin(__builtin_amdgcn_tensor_load_to_lds)
  const u64 ga = (u64)(uintptr_t)gptr;
  const u32 lds_off = (u32)(uintptr_t)lds_ptr;   // low 32 bits of flat = LDS offset
  v4u g0;
  g0.x = 1u;                                      // count=1, user descriptor
  g0.y = lds_off;                                 // lds_addr (bytes)
  g0.z = (u32)ga;                                 // global_addr[31:0]
  g0.w = (u32)((ga >> 32) & 0x1FFFFFFu)           // global_addr[56:32]
       | (2u << 30);                              // type = 2 ("image")
  v8i g1;
  g1[0] = (2 << 16)                               // data_size = 4 bytes
        | (1 << 20)                               // pad_enable
        | (4 << 22)                               // pad_interval code 4 -> every 32 dwords
        | (0 << 25);                              // pad_amount  code 0 -> 1 dword
  g1[1] = (TILE & 0xFFFF) << 16;                  // tensor_dim0 lo16 (=32)
  g1[2] = (TILE & 0xFFFF) << 16;                  // dim0 hi16 (0) | tensor_dim1 lo16 (=32)
  g1[3] = (TILE & 0xFFFF) << 16;                  // dim1 hi16 (0) | tile_dim0 (=32)
  g1[4] = TILE;                                   // tile_dim1 (=32) | tile_dim2 (=0)
  g1[5] = n_stride;                               // tensor_dim0_stride lo32 (= N)
  g1[6] = 0;                                      // stride hi16 | dim1_stride lo16
  g1[7] = 0;
  const v4i zz4 = {0, 0, 0, 0};                   // groups 2/3 unused (2-D tensor)
  const v8i zz8 = {0, 0, 0, 0, 0, 0, 0, 0};
  // clang-23 / therock-10.0 form: 6 args, extra int32x8 group before cpol.
  __builtin_amdgcn_tensor_load_to_lds(g0, g1, zz4, zz4, zz8, 0);
#else
  // Fallback: coalesced manual staging (32 coalesced row loads).
  #pragma unroll
  for (int r = 0; r < TILE; ++r)
    lds_ptr[r * LDS_STRIDE + lane] = gptr[(size_t)r * n_stride + lane];
#endif
}

__device__ __forceinline__ void tdm_wait_le1() {
#if defined(__has_builtin) && __has_builtin(__builtin_amdgcn_s_wait_tensorcnt)
  __builtin_amdgcn_s_wait_tensorcnt(1);
#endif
  __asm__ volatile("" ::: "memory");
}
__device__ __forceinline__ void tdm_wait_le0() {
#if defined(__has_builtin) && __has_builtin(__builtin_amdgcn_s_wait_tensorcnt)
  __builtin_amdgcn_s_wait_tensorcnt(0);
#endif
  __asm__ volatile("" ::: "memory");
}

__global__ __launch_bounds__(TILE) void thomas_fused_kernel(
    const float* __restrict__ alpha,   // (B, N) row-major
    const float* __restrict__ f,       // (N,)  shared by all rows
    float* __restrict__ out,           // (B, N) row-major
    float2* __restrict__ cpdp,         // workspace: (N, B) transposed {cp,dp}
    int Bn, int Nn) {
  __shared__ float smem[2 * TILE_ELEMS];

  const int lane = threadIdx.x;              // 0..31, one wave per block
  const int row0 = blockIdx.x * TILE;        // this wave owns rows row0..row0+31
  const int NT   = Nn / TILE;

  const float* arow = alpha + (size_t)row0 * Nn;   // tile t starts at arow + t*32

  // ---------------- forward sweep ----------------
  tdm_load_tile(&smem[0], arow, Nn, lane);

  float x_prev = 0.f;     // alpha[i-1]  (0 at start -> a_0 = 0, matching the pad)
  float x_cur  = 0.f;     // alpha[i]
  float cp = 0.f, dp = 0.f;

  for (int t = 0; t < NT; ++t) {
    if (t + 1 < NT) {
      tdm_load_tile(&smem[((t + 1) & 1) * TILE_ELEMS],
                    arow + (size_t)(t + 1) * TILE, Nn, lane);
      tdm_wait_le1();                  // tile t (older, in-order) is complete
    } else {
      tdm_wait_le0();                  // last tile complete
    }
    const float* tp = &smem[(t & 1) * TILE_ELEMS] + lane * LDS_STRIDE;
#pragma unroll
    for (int j = 0; j < TILE; ++j) {
      const float x_next = tp[j];      // alpha[i+1], conflict-free (stride 33)
      const int i = t * TILE + j - 1;  // index being solved this step
      if (t > 0 || j > 0) {            // skip only the very first (prime) step
        const float a = x_prev * x_prev;                        // subdiag
        const float b = fmaf(x_cur * x_cur, x_cur, 1.f);        // 1 + alpha^3
        const float c = x_next * (x_next + 2.f);                // superdiag
        const float denom = fmaf(-a, cp, b);
        const float inv   = 1.f / denom;                        // exact divide
        cp = c * inv;
        dp = fmaf(-a, dp, f[i]) * inv;                          // f[i]: scalar load
        cpdp[(size_t)i * Bn + row0 + lane] = make_float2(cp, dp); // coalesced b64
      }
      x_prev = x_cur;
      x_cur  = x_next;
    }
  }
  { // final index i = N-1: c_{N-1} := 0 (pad)
    const int i = Nn - 1;
    const float a = x_prev * x_prev;
    const float b = fmaf(x_cur * x_cur, x_cur, 1.f);
    const float denom = fmaf(-a, cp, b);
    const float inv   = 1.f / denom;
    cp = 0.f;
    dp = fmaf(-a, dp, f[i]) * inv;
    cpdp[(size_t)i * Bn + row0 + lane] = make_float2(cp, dp);
  }

  // ---------------- backward sweep ----------------
  float u_next = 0.f;
  float* otile = smem;                 // reuse buffer 0 as the output tile
  for (int t = NT - 1; t >= 0; --t) {
#pragma unroll
    for (int j = TILE - 1; j >= 0; --j) {
      const int i = t * TILE + j;
      const float2 cd = cpdp[(size_t)i * Bn + row0 + lane];  // coalesced, L2-hot
      const float u = fmaf(-cd.x, u_next, cd.y);             // dp - cp*u_next
      otile[lane * LDS_STRIDE + j] = u;                      // padded -> no conflicts
      u_next = u;
    }
    // Flush the 32x32 tile with coalesced, non-temporal row stores.
#pragma unroll
    for (int r = 0; r < TILE; ++r) {
      const float v = otile[r * LDS_STRIDE + lane];
      __builtin_nontemporal_store(
          v, &out[(size_t)(row0 + r) * Nn + t * TILE + lane]);
    }
  }
}

extern "C" void kernel_launch(void* const* d_in, const int* in_sizes, int n_in,
                              void* d_out, int out_size, void* d_ws, size_t ws_size,
                              hipStream_t stream) {
  (void)n_in; (void)out_size; (void)ws_size;
  const float* alpha = (const float*)d_in[0];   // (B, N) fp32
  const float* f     = (const float*)d_in[1];   // (N,)   fp32
  float* out = (float*)d_out;                   // (B, N) fp32

  const int Nn = in_sizes[1];
  const int Bn = in_sizes[0] / Nn;

  // Needs ws_size >= (size_t)Bn * Nn * sizeof(float2)  (128 MB at 2048x8192).
  float2* cpdp = (float2*)d_ws;

  dim3 grid(Bn / TILE);
  dim3 block(TILE);
  thomas_fused_kernel<<<grid, block, 0, stream>>>(alpha, f, out, cpdp, Bn, Nn);
}